// pose3d_calibration_73830487818417
// MI455X (gfx1250) — compile-verified
//
#include <hip/hip_runtime.h>

#define NJ 15
#define FOCALC 575.0f
#define PXC 320.0f
#define PYC 240.0f

typedef __attribute__((ext_vector_type(2))) float v2f;
typedef __attribute__((ext_vector_type(8))) float v8f;

// Per-frame projection MSE body. With forceinline + full unroll all indices are
// compile-time and r/c/b stay in registers for the vectorized path.
__device__ __forceinline__ float frame_mse(
    const float* __restrict__ px, const float* __restrict__ py,
    const float* __restrict__ pz,
    const float* __restrict__ r,   // 9: R[k*3+c]
    const float* __restrict__ c,   // 3
    const float* __restrict__ b)   // 30: u row then v row
{
  const float r00 = r[0], r01 = r[1], r02 = r[2];
  const float r10 = r[3], r11 = r[4], r12 = r[5];
  const float r20 = r[6], r21 = r[7], r22 = r[8];
  const float c0 = c[0], c1 = c[1], c2 = c[2];

  // Pre-scale u/v columns of R^T by FOCAL (saves 2 muls per joint).
  const float a00 = FOCALC * r00, a10 = FOCALC * r10, a20 = FOCALC * r20;
  const float a01 = FOCALC * r01, a11 = FOCALC * r11, a21 = FOCALC * r21;
  const float t0 = a00 * c0 + a10 * c1 + a20 * c2;
  const float t1 = a01 * c0 + a11 * c1 + a21 * c2;
  const float t2 = r02 * c0 + r12 * c1 + r22 * c2;

  float fs = 0.0f;
#pragma unroll
  for (int j = 0; j < NJ; ++j) {
    const float X = px[j], Y = py[j], Z = pz[j];
    const float pu = fmaf(a00, X, fmaf(a10, Y, fmaf(a20, Z, -t0))); // FOCAL*P0
    const float pv = fmaf(a01, X, fmaf(a11, Y, fmaf(a21, Z, -t1))); // FOCAL*P1
    const float z  = fmaf(r02, X, fmaf(r12, Y, fmaf(r22, Z, -t2)));
    float rz = __builtin_amdgcn_rcpf(z);
    rz = rz * fmaf(-z, rz, 2.0f);            // one Newton step -> ~0.5 ulp
    const float du = fmaf(pu, rz, PXC) - b[j];
    const float dv = fmaf(pv, rz, PYC) - b[NJ + j];
    fs = fmaf(du, du, fs);
    fs = fmaf(dv, dv, fs);
  }
  return fs;
}

// ---------------------------------------------------------------------------
// Kernel 1: streaming projection loss. TWO frames per thread per iteration so
// the dominant bone_2d stream is 16B-aligned: 15x global_load_b128 per pair
// (vs 30x b64), R as 9x b64, C as 3x b64. Consecutive lanes take consecutive
// frame pairs (coalesced 240B-stride streams; all bytes of every line used).
// ---------------------------------------------------------------------------
__global__ __launch_bounds__(256) void proj_partial_kernel(
    const float* __restrict__ pose3d,   // 3*15 (uniform)
    const float* __restrict__ bone2d,   // N*2*15
    const float* __restrict__ Rm,       // N*3*3
    const float* __restrict__ Cm,       // N*3
    float* __restrict__ partials, int N)
{
  // pose3d is uniform across the whole grid: hoist into registers once.
  float px[NJ], py[NJ], pz[NJ];
#pragma unroll
  for (int j = 0; j < NJ; ++j) {
    px[j] = pose3d[j];
    py[j] = pose3d[NJ + j];
    pz[j] = pose3d[2 * NJ + j];
  }

  const int tid    = blockIdx.x * blockDim.x + threadIdx.x;
  const int stride = gridDim.x * blockDim.x;
  const int P      = N >> 1;                 // number of frame pairs

  float acc = 0.0f;
  for (int p = tid; p < P; p += stride) {
    // bone_2d pair block: 60 floats, base 240*p bytes -> 16B aligned
    float bb[60];
    const float4* b4 = reinterpret_cast<const float4*>(bone2d + (size_t)p * 60);
#pragma unroll
    for (int i = 0; i < 15; ++i) {
      float4 t = b4[i];
      bb[4 * i]     = t.x;
      bb[4 * i + 1] = t.y;
      bb[4 * i + 2] = t.z;
      bb[4 * i + 3] = t.w;
    }
    // R pair: 18 floats, base 72*p bytes -> 8B aligned
    float rr[18];
    const float2* r2 = reinterpret_cast<const float2*>(Rm + (size_t)p * 18);
#pragma unroll
    for (int i = 0; i < 9; ++i) {
      float2 t = r2[i];
      rr[2 * i] = t.x; rr[2 * i + 1] = t.y;
    }
    // C pair: 6 floats, base 24*p bytes -> 8B aligned
    float cc[6];
    const float2* c2 = reinterpret_cast<const float2*>(Cm + (size_t)p * 6);
#pragma unroll
    for (int i = 0; i < 3; ++i) {
      float2 t = c2[i];
      cc[2 * i] = t.x; cc[2 * i + 1] = t.y;
    }

#pragma unroll
    for (int f = 0; f < 2; ++f) {
      const float fs = frame_mse(px, py, pz, rr + f * 9, cc + f * 3, bb + f * 30);
      acc = fmaf(fs, (1.0f / 30.0f), acc);   // per-frame MSE
    }
  }

  // Odd-N tail: one designated thread handles the last frame (deterministic).
  if ((N & 1) && tid == 0) {
    const int n = N - 1;
    float bb[30], rr[9], cc[3];
#pragma unroll
    for (int i = 0; i < 30; ++i) bb[i] = bone2d[(size_t)n * 30 + i];
#pragma unroll
    for (int i = 0; i < 9; ++i)  rr[i] = Rm[(size_t)n * 9 + i];
#pragma unroll
    for (int i = 0; i < 3; ++i)  cc[i] = Cm[(size_t)n * 3 + i];
    acc = fmaf(frame_mse(px, py, pz, rr, cc, bb), (1.0f / 30.0f), acc);
  }

  // Deterministic block tree reduction.
  __shared__ float smem[256];
  smem[threadIdx.x] = acc;
  __syncthreads();
#pragma unroll
  for (int off = 128; off > 0; off >>= 1) {
    if (threadIdx.x < off) smem[threadIdx.x] += smem[threadIdx.x + off];
    __syncthreads();
  }
  if (threadIdx.x == 0) partials[blockIdx.x] = smem[0];
}

// ---------------------------------------------------------------------------
// Kernel 2: single block. Fixed-order partial reduction, then a per-wave
// V_WMMA_F32_16X16X4_F32 cross-lane reduction (A = ones(16x4) => D = column
// sums of B; lane values packed into B rows k=0 / k=2; fp32-exact). EXEC is
// all-ones at the WMMA (all 256 threads execute it). Adds symmetry term.
// ---------------------------------------------------------------------------
__global__ __launch_bounds__(256) void final_reduce_kernel(
    const float* __restrict__ partials, int M,
    const float* __restrict__ pose3d,
    const int* __restrict__ lb,   // 6x2
    const int* __restrict__ rb,   // 6x2
    float* __restrict__ out)
{
  // Fixed-stride accumulation: deterministic.
  float acc = 0.0f;
  for (int i = threadIdx.x; i < M; i += 256) acc += partials[i];

  // WMMA reduction of this wave's 32 lane values -> 16 (exact fp32 adds).
  v2f a; a.x = 1.0f; a.y = 1.0f;      // A(16x4) = all ones
  v2f bm; bm.x = acc; bm.y = 0.0f;    // B rows k=0 (lanes 0-15), k=2 (lanes 16-31)
  v8f cm = {};
  cm = __builtin_amdgcn_wmma_f32_16x16x4_f32(
      /*neg_a=*/false, a, /*neg_b=*/false, bm,
      /*c_mod=*/(short)0, cm, /*reuse_a=*/false, /*reuse_b=*/false);
  float x = cm[0];                    // lane n: acc[n%16] + acc[n%16 + 16]
  x += __shfl_xor(x, 8, 32);
  x += __shfl_xor(x, 4, 32);
  x += __shfl_xor(x, 2, 32);
  x += __shfl_xor(x, 1, 32);          // every lane: wave total

  __shared__ float wsum[8];
  const int wid  = threadIdx.x >> 5;
  const int lane = threadIdx.x & 31;
  if (lane == 0) wsum[wid] = x;
  __syncthreads();

  if (threadIdx.x == 0) {
    float proj = 0.0f;
#pragma unroll
    for (int w = 0; w < 8; ++w) proj += wsum[w];

    // Symmetry term (tiny, fixed order).
    float sym = 0.0f;
#pragma unroll
    for (int i = 0; i < 6; ++i) {
      const int l0 = lb[2 * i], l1 = lb[2 * i + 1];
      const int r0 = rb[2 * i], r1 = rb[2 * i + 1];
      float ll = 0.0f, lr = 0.0f;
#pragma unroll
      for (int cidx = 0; cidx < 3; ++cidx) {
        float d = pose3d[cidx * NJ + l0] - pose3d[cidx * NJ + l1];
        ll = fmaf(d, d, ll);
        d = pose3d[cidx * NJ + r0] - pose3d[cidx * NJ + r1];
        lr = fmaf(d, d, lr);
      }
      const float e = ll - lr;
      sym = fmaf(e, e, sym);
    }
    sym *= (1.0f / 6.0f);

    out[0] = sym + proj;   // W_SYM = W_PROJ = 1
  }
}

extern "C" void kernel_launch(void* const* d_in, const int* in_sizes, int n_in,
                              void* d_out, int out_size, void* d_ws, size_t ws_size,
                              hipStream_t stream) {
  (void)n_in; (void)out_size;
  const float* pose3d = (const float*)d_in[0];
  const float* bone2d = (const float*)d_in[1];
  const float* Rm     = (const float*)d_in[2];
  const float* Cm     = (const float*)d_in[3];
  const int*   lb     = (const int*)d_in[4];
  const int*   rb     = (const int*)d_in[5];

  const int N = in_sizes[1] / 30;

  int blocks = 1024;
  const int max_blocks = (int)(ws_size / sizeof(float));
  if (blocks > max_blocks) blocks = max_blocks > 0 ? max_blocks : 1;

  float* partials = (float*)d_ws;

  proj_partial_kernel<<<blocks, 256, 0, stream>>>(pose3d, bone2d, Rm, Cm,
                                                  partials, N);
  final_reduce_kernel<<<1, 256, 0, stream>>>(partials, blocks, pose3d, lb, rb,
                                             (float*)d_out);
}